// ChannelScorer_Distributed_39024072851482
// MI455X (gfx1250) — compile-verified
//
#include <hip/hip_runtime.h>
#include <hip/hip_bf16.h>

// ---------------- problem constants ----------------
#define B_      256
#define M_      64
#define T_      1200
#define FT_     10          // conv filters
#define K_      17          // conv kernel taps
#define POOLK_  75
#define POOLS_  15
#define TP_     76          // pooled length
#define FEAT_   (FT_ * TP_) // 760
#define NCHUNK_ 80          // T_/POOLS_ : 15-wide chunk sums
#define NTILE_  75          // T_/16 WMMA time tiles
#define EPS_    1e-10f

// ---------------- LDS layout (per wave) ----------------
#define XP_LEN     1224                 // x padded: t=-8..1215, tail zeroed (max read 1218)
#define SQ_STRIDE  20                   // 16x16 tile staging, stride 20 -> conflict-free halves
#define SQ_LEN     (16 * SQ_STRIDE)     // 320
#define S_LEN      (FT_ * NCHUNK_)      // 800
#define WAVE_LDS_PAD 2368               // 1224+320+800 = 2344, padded
#define WAVES      4                    // 4 waves (rows) per block -> ~38 KB LDS

typedef float v2f __attribute__((ext_vector_type(2)));
typedef float v8f __attribute__((ext_vector_type(8)));

// Wave-private LDS producer->consumer ordering (no workgroup barrier needed):
// drain DScnt and fence the scheduler.
__device__ __forceinline__ void wave_sync_lds() {
    __builtin_amdgcn_wave_barrier();
    asm volatile("s_wait_dscnt 0" ::: "memory");
    __builtin_amdgcn_wave_barrier();
}

__global__ __launch_bounds__(WAVES * 32)
void channel_scorer_kernel(const float* __restrict__ x,
                           const float* __restrict__ conv_w,   // (10,1,17)
                           const float* __restrict__ conv_b,   // (10)
                           const float* __restrict__ lin_w,    // (64,760)
                           const float* __restrict__ lin_b,    // (64)
                           float* __restrict__ out)            // (256,64,1)
{
    __shared__ __attribute__((aligned(16))) float lds[WAVES][WAVE_LDS_PAD];

    const int wave = threadIdx.x >> 5;
    const int lane = threadIdx.x & 31;
    const int row  = blockIdx.x * WAVES + wave;   // row = b*M + m, 0..16383
    const int m    = row & (M_ - 1);

    float* xp = &lds[wave][0];                    // padded signal, xp[8+t] = x[t]
    float* sq = &lds[wave][XP_LEN];               // squared conv tile staging [16][20]
    float* S  = &lds[wave][XP_LEN + SQ_LEN];      // chunk sums [10][80]

    // -------- async DMA the x row straight into LDS (no VGPR round trip) --------
    // GLOBAL_LOAD_ASYNC_TO_LDS_B128, GV mode: 64-bit address in a VGPR pair,
    // SADDR = off. VDST VGPR holds the LDS byte offset (flat-LDS low 32 bits).
    {
        const float* grow = x + (size_t)row * T_;
        for (int i = lane; i < T_ / 4; i += 32) {             // 300 x 16B transfers
            unsigned     ldsoff = (unsigned)(uintptr_t)(&xp[8 + 4 * i]);
            const float* gaddr  = grow + 4 * i;               // 16B aligned
            asm volatile("global_load_async_to_lds_b128 %0, %1, off"
                         :: "v"(ldsoff), "v"(gaddr)
                         : "memory");
        }
    }

    // -------- lane roles for the 16x16x4 f32 WMMA --------
    // A (16x4 f32): lanes 0-15 -> M=lane, VGPR{0,1}=K{0,1}; lanes 16-31 -> K{2,3}
    // B (4x16 f32): lanes 0-15 -> N=lane, VGPR{0,1}=K{0,1}; lanes 16-31 -> K{2,3}
    // C (16x16 f32): VGPR r -> M = r + 8*(lane>=16), N = lane&15
    const int nIdx  = lane & 15;            // filter row for A, time col for B/C
    const int kk0   = (lane >> 4) * 2;      // K sub-offset for this half-wave
    const int fBase = (lane >> 4) * 8;      // C-tile filter base for this half-wave

    // A operands: 5 K-chunks of the zero-padded 16x20 weight matrix (held in regs);
    // these global loads overlap the async LDS fill.
    v2f a[5];
#pragma unroll
    for (int c = 0; c < 5; ++c) {
        const int k0 = 4 * c + kk0;
        float a0 = 0.0f, a1 = 0.0f;
        if (nIdx < FT_) {
            if (k0     < K_) a0 = conv_w[nIdx * K_ + k0];
            if (k0 + 1 < K_) a1 = conv_w[nIdx * K_ + k0 + 1];
        }
        a[c][0] = a0; a[c][1] = a1;
    }

    // Per-lane conv bias for the 8 C rows this lane owns
    float biasr[8];
#pragma unroll
    for (int r = 0; r < 8; ++r) {
        const int f = fBase + r;
        biasr[r] = (f < FT_) ? conv_b[f] : 0.0f;
    }

    const float* lw = lin_w + m * FEAT_;
    __builtin_prefetch(lw, 0, 1);               // global_prefetch_b8 -> warm L2 for final dot

    // Zero pads + chunk sums while the async fill is in flight
    if (lane < 8)  xp[lane] = 0.0f;                       // t = -8..-1
    if (lane < 16) xp[8 + T_ + lane] = 0.0f;              // t = 1200..1215 (covers K pad reads)
    for (int i = lane; i < S_LEN; i += 32) S[i] = 0.0f;   // zero chunk sums

    asm volatile("s_wait_asynccnt 0" ::: "memory");       // async DMA -> LDS complete
    wave_sync_lds();                                      // + drain DS stores

    // -------- conv (WMMA) + square + chunk-sum pooling, 75 tiles of 16 times --------
    for (int tt = 0; tt < NTILE_; ++tt) {
        const int t0 = tt * 16;

        v8f acc = {0.f, 0.f, 0.f, 0.f, 0.f, 0.f, 0.f, 0.f};
        const int bbase = t0 + nIdx + kk0;      // xp[t + k] = x[t + k - 8]
#pragma unroll
        for (int c = 0; c < 5; ++c) {
            v2f b;
            b[0] = xp[bbase + 4 * c];
            b[1] = xp[bbase + 4 * c + 1];
            acc = __builtin_amdgcn_wmma_f32_16x16x4_f32(
                      false, a[c], false, b, (short)0, acc, false, false);
        }

        // (y + bias)^2 -> staging tile
#pragma unroll
        for (int r = 0; r < 8; ++r) {
            float v = acc[r] + biasr[r];
            sq[(fBase + r) * SQ_STRIDE + nIdx] = v * v;
        }
        wave_sync_lds();

        // each 16-wide tile spans exactly 2 pooling chunks (chunk = 15 samples)
        const int q0    = t0 / POOLS_;
        const int split = POOLS_ * (q0 + 1) - t0;     // in [1,15]
        const int g     = lane >> 4;                  // 0 -> chunk q0, 1 -> chunk q0+1
        const int f     = lane & 15;
        if (f < FT_) {
            const int nlo = g ? split : 0;
            const int nhi = g ? 16 : split;
            float s = 0.0f;
            for (int n = nlo; n < nhi; ++n) s += sq[f * SQ_STRIDE + n];
            S[f * NCHUNK_ + q0 + g] += s;             // wave-private RMW
        }
        wave_sync_lds();
    }

    // -------- pooled -> log -> per-channel dot (pooled[p] = sum of 5 chunk sums) --------
    float dot = 0.0f;
    for (int e = lane; e < FEAT_; e += 32) {          // feature index = f*76 + p
        const int f = e / TP_;
        const int p = e - f * TP_;
        const float* Sf = S + f * NCHUNK_ + p;
        float pooled = (Sf[0] + Sf[1] + Sf[2] + Sf[3] + Sf[4]) * (1.0f / POOLK_);
        pooled = fmaxf(pooled, EPS_);
        dot += __logf(pooled) * lw[e];
    }
#pragma unroll
    for (int off = 16; off > 0; off >>= 1)
        dot += __shfl_xor(dot, off, 32);
    if (lane == 0) out[row] = dot + lin_b[m];
}

extern "C" void kernel_launch(void* const* d_in, const int* in_sizes, int n_in,
                              void* d_out, int out_size, void* d_ws, size_t ws_size,
                              hipStream_t stream) {
    (void)in_sizes; (void)n_in; (void)out_size; (void)d_ws; (void)ws_size;
    const float* x      = (const float*)d_in[0];
    const float* conv_w = (const float*)d_in[1];
    const float* conv_b = (const float*)d_in[2];
    const float* lin_w  = (const float*)d_in[3];
    const float* lin_b  = (const float*)d_in[4];
    float* out          = (float*)d_out;

    dim3 grid((B_ * M_) / WAVES);   // 4096 blocks
    dim3 block(WAVES * 32);         // 128 threads = 4 waves, 1 wave per row
    hipLaunchKernelGGL(channel_scorer_kernel, grid, block, 0, stream,
                       x, conv_w, conv_b, lin_w, lin_b, out);
}